// HashEmbedder2D_87325275062764
// MI455X (gfx1250) — compile-verified
//
#include <hip/hip_runtime.h>
#include <stdint.h>
#include <stddef.h>

// ---------------- compile-time replication of the reference constants ----------
#define NLEV 16
#define NPTS (1u << 21)          // 2,097,152 points
#define BLK 256                  // 8 wave32 waves
#define NBLOCKS 2048
#define CHUNKS (NPTS / BLK)      // 8192 chunks of 256 points
#define NSTAGED 4                // levels 0..3 cached in LDS (27,232 B)
#define OUT_STRIDE 36            // LDS transpose row stride (floats), 144B = 16B-aligned

typedef float f32x4 __attribute__((ext_vector_type(4)));

static constexpr int      RES[NLEV] = {16, 22, 30, 42, 58, 80, 111, 153,
                                       212, 294, 406, 561, 776, 1072, 1482, 2048};
static constexpr uint32_t TS[NLEV]  = {256, 484, 900, 1764, 3364, 6400, 12321, 23409,
                                       44944, 86436, 164836, 314721,
                                       524288, 524288, 524288, 524288};
// float2-unit offsets of staged tables inside s_tab (each *8B is 16B aligned)
static constexpr uint32_t SOFF[NSTAGED] = {0, 256, 740, 1640};
#define STAGED_F2 3404           // total staged float2 entries (27,232 bytes)

#define P1C 2654435761ull
#define P2C 805459861ull

struct Params {
    const float2* __restrict__ x;
    const float2* __restrict__ tab[NLEV];
    float*        __restrict__ out;
};

// CDNA5 async copy: 16B global -> LDS, tracked on ASYNCcnt (per-lane addresses).
__device__ __forceinline__ void async_copy16(uint32_t lds_byte_off, uint64_t gaddr) {
    asm volatile("global_load_async_to_lds_b128 %0, %1, off"
                 :: "v"(lds_byte_off), "v"(gaddr)
                 : "memory");
}

// h % ts for h < 2^43, ts < 2^20, using only one mad_u64_u32 + 32-bit magic urem.
// Exactness: h = hi*2^32+lo == hi*C+lo = s (mod ts), s < 2^31+2^32;
// s = s_hi*2^32+s_lo == s_hi*C+s_lo = t (mod ts); if s_hi==1 then s_lo < 2^31,
// so t = s_lo + s_hi*C < 2^31 + 2^20 never wraps 32 bits. Then 32-bit t % ts.
__device__ __forceinline__ uint32_t mod_ts(uint64_t h, uint32_t ts, uint32_t C) {
    const uint32_t hi = (uint32_t)(h >> 32);
    const uint32_t lo = (uint32_t)h;
    const uint64_t s  = (uint64_t)hi * C + lo;
    const uint32_t t  = (uint32_t)s + (uint32_t)(s >> 32) * C;
    return t % ts;   // ts is a constant at every (unrolled) call site
}

__global__ __launch_bounds__(BLK)
void hashenc2d_kernel(Params p) {
    __shared__ __align__(16) float2 s_tab[STAGED_F2];           // 27,232 B
    __shared__ __align__(16) float  s_out[BLK * OUT_STRIDE];    // 36,864 B

    const int tid = threadIdx.x;

    // ---- stage the hot small tables into LDS via CDNA5 async-to-LDS ----
    {
        const uint32_t lds0 = (uint32_t)(uintptr_t)(&s_tab[0]);
#pragma unroll
        for (int l = 0; l < NSTAGED; ++l) {
            const uint32_t nvec = TS[l] / 2;  // # of 16-byte vectors (TS*8B/16)
            const uint64_t g    = (uint64_t)(uintptr_t)p.tab[l];
            for (uint32_t i = (uint32_t)tid; i < nvec; i += BLK) {
                async_copy16(lds0 + SOFF[l] * 8u + i * 16u, g + (uint64_t)i * 16u);
            }
        }
        asm volatile("s_wait_asynccnt 0" ::: "memory");
    }
    __syncthreads();

    for (uint32_t c = blockIdx.x; c < CHUNKS; c += NBLOCKS) {
        const uint32_t pidx = c * BLK + (uint32_t)tid;
        const float2 xy = p.x[pidx];
        const float xn = fminf(fmaxf((xy.x + 1.0f) * 0.5f, 0.0f), 1.0f);
        const float yn = fminf(fmaxf((xy.y + 1.0f) * 0.5f, 0.0f), 1.0f);

        float feat[2 * NLEV];
#pragma unroll
        for (int l = 0; l < NLEV; ++l) {
            const float rf  = (float)RES[l];
            const int rmax  = RES[l] - 1;
            const uint32_t ts = TS[l];
            const uint32_t C  = (uint32_t)(0x100000000ull % TS[l]);  // constexpr folds
            const float xs  = xn * rf;
            const float ys  = yn * rf;
            const float xf  = floorf(xs);
            const float yf  = floorf(ys);
            const float wx  = xs - xf;
            const float wy  = ys - yf;
            const int xi = (int)xf;
            const int yi = (int)yf;
            // reference: clip(int(xf){+1}, 0, res-1) -> single v_med3_i32 each
            const uint32_t x0u = (uint32_t)min(max(xi, 0), rmax);
            const uint32_t x1u = (uint32_t)min(max(xi + 1, 0), rmax);
            const uint32_t y0u = (uint32_t)min(max(yi, 0), rmax);
            const uint32_t y1u = (uint32_t)min(max(yi + 1, 0), rmax);

            // exact int64 hash semantics of the reference
            const uint64_t hx0 = (uint64_t)x0u * P1C;
            const uint64_t hx1 = (uint64_t)x1u * P1C;
            const uint64_t hy0 = (uint64_t)y0u * P2C;
            const uint64_t hy1 = (uint64_t)y1u * P2C;
            const uint32_t i00 = mod_ts(hx0 ^ hy0, ts, C);
            const uint32_t i10 = mod_ts(hx1 ^ hy0, ts, C);
            const uint32_t i01 = mod_ts(hx0 ^ hy1, ts, C);
            const uint32_t i11 = mod_ts(hx1 ^ hy1, ts, C);

            float2 e00, e10, e01, e11;
            if (l < NSTAGED) {
                const float2* t = &s_tab[SOFF[l]];
                e00 = t[i00]; e10 = t[i10]; e01 = t[i01]; e11 = t[i11];
            } else {
                const float2* __restrict__ t = p.tab[l];
                e00 = t[i00]; e10 = t[i10]; e01 = t[i01]; e11 = t[i11];
            }

            const float f0x = e00.x + (e10.x - e00.x) * wx;
            const float f0y = e00.y + (e10.y - e00.y) * wx;
            const float f1x = e01.x + (e11.x - e01.x) * wx;
            const float f1y = e01.y + (e11.y - e01.y) * wx;
            feat[2 * l + 0] = f0x + (f1x - f0x) * wy;
            feat[2 * l + 1] = f0y + (f1y - f0y) * wy;
        }

        // ---- LDS transpose: per-point 128B rows -> fully coalesced b128 stores ----
#pragma unroll
        for (int k = 0; k < 8; ++k) {
            f32x4 v;
            v.x = feat[4 * k + 0];
            v.y = feat[4 * k + 1];
            v.z = feat[4 * k + 2];
            v.w = feat[4 * k + 3];
            *(f32x4*)&s_out[tid * OUT_STRIDE + k * 4] = v;
        }
        __syncthreads();

        const size_t base = (size_t)c * (BLK * 2 * NLEV);
#pragma unroll
        for (int k = 0; k < 8; ++k) {
            const int w  = k * 1024 + tid * 4;     // flat float index in this chunk
            const int pl = w >> 5;                 // point within chunk
            const int j  = w & 31;                 // feature index (multiple of 4)
            const f32x4 v = *(const f32x4*)&s_out[pl * OUT_STRIDE + j];
            // streaming output: non-temporal so 256MB stream doesn't evict tables
            __builtin_nontemporal_store(v, (f32x4*)&p.out[base + (size_t)w]);
        }
        __syncthreads();
    }
}

extern "C" void kernel_launch(void* const* d_in, const int* in_sizes, int n_in,
                              void* d_out, int out_size, void* d_ws, size_t ws_size,
                              hipStream_t stream) {
    (void)in_sizes; (void)n_in; (void)out_size; (void)d_ws; (void)ws_size;
    Params p;
    p.x = (const float2*)d_in[0];
    for (int i = 0; i < NLEV; ++i) p.tab[i] = (const float2*)d_in[1 + i];
    p.out = (float*)d_out;
    hipLaunchKernelGGL(hashenc2d_kernel, dim3(NBLOCKS), dim3(BLK), 0, stream, p);
}